// Anchors_56435870269539
// MI455X (gfx1250) — compile-verified
//
#include <hip/hip_runtime.h>
#include <hip/hip_bf16.h>
#include <math.h>

#ifndef __has_builtin
#define __has_builtin(x) 0
#endif

#if __has_builtin(__builtin_amdgcn_global_load_async_to_lds_b128)
#define USE_ASYNC_LDS 1
#else
#define USE_ASYNC_LDS 0
#endif

#define N_ANCH 48960
#define BATCH  8
#define MGT    64
#define BN     (BATCH * N_ANCH)

// The async-to-LDS builtin takes pointers to 4xi32 vectors in explicit
// address spaces: (global v4i*, local v4i*, imm offset, imm cpol).
typedef int v4i __attribute__((ext_vector_type(4)));
typedef __attribute__((address_space(1))) v4i v4i_as1;
typedef __attribute__((address_space(3))) v4i v4i_as3;

__device__ __forceinline__ void async_wait0() {
#if USE_ASYNC_LDS
#if __has_builtin(__builtin_amdgcn_s_wait_asynccnt)
  __builtin_amdgcn_s_wait_asynccnt(0);
#else
  asm volatile("s_wait_asynccnt 0" ::: "memory");
#endif
#endif
}

// Stage one batch's 64 GT boxes (1KB) into LDS via the gfx1250 async
// global->LDS DMA path (ASYNCcnt), bypassing VGPRs entirely.
__device__ __forceinline__ void stage_gt(const float* __restrict__ gt_boxes,
                                         int b, float4* s_gt) {
  if (threadIdx.x < MGT) {
    const float4* g = (const float4*)gt_boxes + b * MGT + (int)threadIdx.x;
#if USE_ASYNC_LDS
    __builtin_amdgcn_global_load_async_to_lds_b128(
        (v4i_as1*)g, (v4i_as3*)&s_gt[threadIdx.x], 0, 0);
#else
    s_gt[threadIdx.x] = *g;
#endif
  }
}

// (w, h) for anchor shape (lvl, a) through f64, cast f32 — matches numpy
// float64 -> float32 in the reference exactly.
__device__ __forceinline__ float2 anchor_wh(int lvl, int a) {
  int size = 32 << lvl;
  int s = a % 3, r = a / 3;
  double scl = (s == 0) ? 1.0
             : ((s == 1) ? 1.2599210498948731648      /* 2^(1/3) */
                         : 1.5874010519681994748);    /* 2^(2/3) */
  double sq  = (r == 0) ? 0.70710678118654752440      /* sqrt(0.5) */
             : ((r == 1) ? 1.0 : 1.41421356237309504880); /* sqrt(2) */
  double base = (double)size * scl;
  return make_float2((float)(base * sq), (float)(base / sq));
}

// Full procedural anchor (cx, cy, w, h): used where it's evaluated once per
// thread (K1/K3). cx/cy are exact in f32 (power-of-2 stride), w/h via f64.
__device__ __forceinline__ float4 anchor_of(int i) {
  int off, ls, lvl;
  if (i < 36864)      { lvl = 0; off = 0;     ls = 6; }   // 64x64, stride 8,  size 32
  else if (i < 46080) { lvl = 1; off = 36864; ls = 5; }   // 32x32, stride 16, size 64
  else if (i < 48384) { lvl = 2; off = 46080; ls = 4; }   // 16x16, stride 32, size 128
  else                { lvl = 3; off = 48384; ls = 3; }   //  8x8,  stride 64, size 256
  int li  = i - off;
  int a   = li % 9;
  int pix = li / 9;
  int x = pix & ((1 << ls) - 1);
  int y = pix >> ls;
  float stride = (float)(8 << lvl);
  float2 wh = anchor_wh(lvl, a);
  float4 o;
  o.x = ((float)x + 0.5f) * stride;   // exact, == f64 result rounded
  o.y = ((float)y + 0.5f) * stride;
  o.z = wh.x;
  o.w = wh.y;
  return o;
}

__device__ __forceinline__ float iou_eval(float ax1, float ay1, float ax2, float ay2,
                                          float area_a, float4 g) {
  float lx = fmaxf(ax1, g.x), ly = fmaxf(ay1, g.y);
  float rx = fminf(ax2, g.z), ry = fminf(ay2, g.w);
  float iw = fmaxf(rx - lx, 0.0f), ih = fmaxf(ry - ly, 0.0f);
  float inter = iw * ih;
  float ag = (g.z - g.x) * (g.w - g.y);
  return inter / (area_a + ag - inter);
}

// K1: per (b, anchor) -> best IoU + argmax over the 64 GT boxes.
__global__ void k_best_per_anchor(const float* __restrict__ gt_boxes,
                                  float* __restrict__ best_t,
                                  int* __restrict__ best_i) {
  const int b = blockIdx.y;
  const int i = blockIdx.x * blockDim.x + threadIdx.x;
  __shared__ float4 s_gt[MGT];
  stage_gt(gt_boxes, b, s_gt);
  async_wait0();
  __syncthreads();
  if (i >= N_ANCH) return;

  float4 an = anchor_of(i);
  float ax1 = an.x - an.z * 0.5f, ay1 = an.y - an.w * 0.5f;
  float ax2 = an.x + an.z * 0.5f, ay2 = an.y + an.w * 0.5f;
  float area_a = (ax2 - ax1) * (ay2 - ay1);

  float best = -1.0f; int bidx = 0;
#pragma unroll 4
  for (int m = 0; m < MGT; ++m) {
    float iou = iou_eval(ax1, ay1, ax2, ay2, area_a, s_gt[m]);
    if (iou > best) { best = iou; bidx = m; }   // strict '>' => first-max (jnp.argmax)
  }
  best_t[b * N_ANCH + i] = best;
  best_i[b * N_ANCH + i] = bidx;
}

// K2: per (b, gt) -> argmax IoU over all 48960 anchors (hot kernel: 25M
// anchor evals). Anchor shapes come from a 36-entry LDS table (f64 math done
// once per block); per-level sub-loops keep level constants in SGPRs and the
// inner loop pure f32/int. Packed u64 key (iou_bits<<32)|~idx gives max-IoU
// with smallest-index tie-break; LDS tree reduce; deterministic, no atomics.
__global__ void k_best_per_gt(const float* __restrict__ gt_boxes,
                              unsigned* __restrict__ bestp) {
  const int m = blockIdx.x, b = blockIdx.y;
  const int tid = threadIdx.x;

  __shared__ float2 s_wh[36];           // [lvl*9 + a] -> (w, h)
  if (tid < 36) s_wh[tid] = anchor_wh(tid / 9, tid % 9);
  __syncthreads();

  const float4 g = ((const float4*)gt_boxes)[b * MGT + m];

  unsigned long long key = 0ULL;
  constexpr int offs[5] = {0, 36864, 46080, 48384, 48960};
#pragma unroll
  for (int lvl = 0; lvl < 4; ++lvl) {
    const int   ls     = 6 - lvl;
    const float stride = (float)(8 << lvl);
    for (int i = offs[lvl] + tid; i < offs[lvl + 1]; i += 256) {
      int li  = i - offs[lvl];
      int a   = li % 9;
      int pix = li / 9;
      int x = pix & ((1 << ls) - 1);
      int y = pix >> ls;
      float2 wh = s_wh[lvl * 9 + a];
      float cx = ((float)x + 0.5f) * stride;
      float cy = ((float)y + 0.5f) * stride;
      float ax1 = cx - wh.x * 0.5f, ay1 = cy - wh.y * 0.5f;
      float ax2 = cx + wh.x * 0.5f, ay2 = cy + wh.y * 0.5f;
      float area_a = (ax2 - ax1) * (ay2 - ay1);
      float iou = iou_eval(ax1, ay1, ax2, ay2, area_a, g);
      unsigned long long k =
          ((unsigned long long)__float_as_uint(iou) << 32) | (unsigned)(~i);
      if (k > key) key = k;
    }
  }

  __shared__ unsigned long long red[256];
  red[tid] = key;
  __syncthreads();
  for (int s = 128; s > 0; s >>= 1) {
    if (tid < s) { if (red[tid + s] > red[tid]) red[tid] = red[tid + s]; }
    __syncthreads();
  }
  if (tid == 0) bestp[b * MGT + m] = ~(unsigned)(red[0] & 0xFFFFFFFFULL);
}

// K3: override loop (ascending m, later m wins == fori_loop), labels + deltas.
__global__ void k_encode(const float* __restrict__ gt_boxes,
                         const int* __restrict__ gt_labels,
                         const float* __restrict__ best_t,
                         const int* __restrict__ best_i,
                         const unsigned* __restrict__ bestp,
                         float* __restrict__ out) {
  const int b = blockIdx.y;
  const int i = blockIdx.x * blockDim.x + threadIdx.x;
  __shared__ float4   s_gt[MGT];
  __shared__ int      s_lab[MGT];
  __shared__ unsigned s_bp[MGT];
  stage_gt(gt_boxes, b, s_gt);
  if (threadIdx.x < MGT) {
    s_lab[threadIdx.x] = gt_labels[b * MGT + threadIdx.x];
    s_bp[threadIdx.x]  = bestp[b * MGT + threadIdx.x];
  }
  async_wait0();
  __syncthreads();
  if (i >= N_ANCH) return;

  float val = best_t[b * N_ANCH + i];
  int   idx = best_i[b * N_ANCH + i];
#pragma unroll 8
  for (int m = 0; m < MGT; ++m) {        // LDS broadcast reads
    if (s_bp[m] == (unsigned)i) { idx = m; val = 2.0f; }
  }

  int lab = s_lab[idx];
  if (val > 0.4f && val < 0.5f) lab = -1;   // ignore band
  if (val < 0.4f)               lab = 0;    // background (applied last, as ref)

  float4 g  = s_gt[idx];
  float4 an = anchor_of(i);
  float bcx = (g.x + g.z) * 0.5f, bcy = (g.y + g.w) * 0.5f;
  float bw  = g.z - g.x,          bh  = g.w - g.y;
  float dx = ((bcx - an.x) / an.z) / 0.1f;
  float dy = ((bcy - an.y) / an.w) / 0.1f;
  float dw = logf(bw / an.z) / 0.2f;
  float dh = logf(bh / an.w) / 0.2f;

  ((float4*)out)[b * N_ANCH + i] = make_float4(dx, dy, dw, dh);
  out[(size_t)BN * 4 + (size_t)b * N_ANCH + i] = (float)lab;
}

extern "C" void kernel_launch(void* const* d_in, const int* in_sizes, int n_in,
                              void* d_out, int out_size, void* d_ws, size_t ws_size,
                              hipStream_t stream) {
  (void)in_sizes; (void)n_in; (void)out_size; (void)ws_size;
  // inputs: f3,f4,f5,f6 (shapes hardcoded; contents unused), gt_boxes, gt_labels
  const float* gt_boxes  = (const float*)d_in[4];
  const int*   gt_labels = (const int*)d_in[5];

  float*    best_t = (float*)d_ws;                               // BN floats
  int*      best_i = (int*)((char*)d_ws + (size_t)BN * 4);       // BN ints
  unsigned* bestp  = (unsigned*)((char*)d_ws + (size_t)BN * 8);  // B*M uints

  dim3 blk(256);
  dim3 grdA((N_ANCH + 255) / 256, BATCH);
  dim3 grdB(MGT, BATCH);
  k_best_per_anchor<<<grdA, blk, 0, stream>>>(gt_boxes, best_t, best_i);
  k_best_per_gt<<<grdB, blk, 0, stream>>>(gt_boxes, bestp);
  k_encode<<<grdA, blk, 0, stream>>>(gt_boxes, gt_labels, best_t, best_i, bestp,
                                     (float*)d_out);
}